// SSDLoss_3418793967629
// MI455X (gfx1250) — compile-verified
//
#include <hip/hip_runtime.h>
#include <hip/hip_bf16.h>

typedef __attribute__((ext_vector_type(2))) float v2f;
typedef __attribute__((ext_vector_type(8))) float v8f;

#define B_        64
#define P_        24576
#define C_        81
#define BLK_ANCH  128
#define THREADS1  128

// Async global->LDS staging path (gfx1250): gated so the file always compiles.
#if __has_builtin(__builtin_amdgcn_global_load_async_to_lds_b128) && \
    __has_builtin(__builtin_amdgcn_s_wait_asynccnt)
#define SSD_ASYNC_STAGE 1
#else
#define SSD_ASYNC_STAGE 0
#endif

#if SSD_ASYNC_STAGE
typedef int v4i_vs __attribute__((vector_size(16)));                 // int4, matches builtin pointee
typedef __attribute__((address_space(1))) v4i_vs* gb128_t;           // global int4*  (64-bit)
typedef __attribute__((address_space(3))) v4i_vs* sb128_t;           // LDS    int4*  (32-bit)
#endif

// Workspace layout (floats): W[0] = scalar loss accumulator,
// ((int*)W)[16..79] = num_pos per batch row, W[128 ...] = ce_neg [B*P].

__global__ void ssd_zero(float* __restrict__ W) {
    W[threadIdx.x] = 0.0f;   // zeros both the f32 acc and the int num_pos region
}

__global__ __launch_bounds__(THREADS1)
void ssd_main(const float* __restrict__ locP, const float* __restrict__ clsP,
              const float* __restrict__ locT, const int* __restrict__ clsT,
              float* __restrict__ W)
{
    __shared__ __align__(16) float scls[BLK_ANCH * C_ + 4]; // +4 slack for padded class reads
    __shared__ float ssum[BLK_ANCH];
    __shared__ float sredf[THREADS1 / 32];
    __shared__ int   sredi[THREADS1 / 32];

    const int tid = threadIdx.x;
    const long long blockAnchor0 = (long long)blockIdx.x * BLK_ANCH;

    // ---- stage one contiguous 128-anchor x 81-class block into LDS ----
#if SSD_ASYNC_STAGE
    {
        const unsigned long long srcB = (unsigned long long)(const void*)(clsP + blockAnchor0 * C_);
        const unsigned ldsBase = (unsigned)(unsigned long long)(const void*)scls; // low 32 bits = LDS byte offset
        const int n4 = (BLK_ANCH * C_) / 4;   // 2592 x b128, 16B-aligned base
        for (int i = tid; i < n4; i += THREADS1) {
            gb128_t g = (gb128_t)(srcB + (unsigned long long)i * 16ull);
            sb128_t l = (sb128_t)(ldsBase + (unsigned)i * 16u);
            __builtin_amdgcn_global_load_async_to_lds_b128(g, l, 0, 0);
        }
        __builtin_amdgcn_s_wait_asynccnt(0);  // this wave's async transfers complete
    }
#else
    {
        const float4* __restrict__ src = (const float4*)(clsP + blockAnchor0 * C_);
        float4* dst = (float4*)scls;
        const int n4 = (BLK_ANCH * C_) / 4;   // 2592, 16B-aligned base
        for (int i = tid; i < n4; i += THREADS1) dst[i] = src[i];
    }
#endif
    __syncthreads();

    // ---- per-anchor max over 81 classes (one anchor per thread, conflict-free stride 81) ----
    float mx = -3.402823466e38f;
    {
        const float* row = scls + tid * C_;
        #pragma unroll 3
        for (int c = 0; c < C_; ++c) mx = fmaxf(mx, row[c]);
    }

    // ---- WMMA sum of exp(x - max): two 16-anchor tiles per wave ----
    const int lane     = tid & 31;
    const int waveBase = tid & ~31;
    const int m15      = lane & 15;
    const int kOff     = (lane >> 4) << 1;   // lanes 0-15 -> K=0,1 ; lanes 16-31 -> K=2,3

    for (int t = 0; t < 2; ++t) {
        const int tileBase = waveBase + t * 16;
        const float tmax = __shfl(mx, t * 16 + m15, 32);   // max of this lane's WMMA-row anchor
        const float* row = scls + (tileBase + m15) * C_;
        v8f c = {};
        v2f bones = {1.0f, 1.0f};                          // B = ones(4x16) -> row sums
        #pragma unroll
        for (int kc = 0; kc < 21; ++kc) {                  // 21*4 = 84 >= 81 classes (padded)
            const int c0 = kc * 4 + kOff;
            const float v0 = row[c0];                      // in-bounds via +4 slack
            const float v1 = row[c0 + 1];
            v2f a;
            a.x = (c0     < C_) ? __expf(v0 - tmax) : 0.0f;
            a.y = (c0 + 1 < C_) ? __expf(v1 - tmax) : 0.0f;
            // D = A(16x4) * ones(4x16) + C  ->  per-anchor row sums in every column
            c = __builtin_amdgcn_wmma_f32_16x16x4_f32(false, a, false, bones,
                                                      (short)0, c, false, false);
        }
        // D layout: VGPR j -> lanes 0-15: M=j, lanes 16-31: M=j+8
        if (lane == 0) {
            #pragma unroll
            for (int j = 0; j < 8; ++j) ssum[tileBase + j] = c[j];
        } else if (lane == 16) {
            #pragma unroll
            for (int j = 0; j < 8; ++j) ssum[tileBase + 8 + j] = c[j];
        }
    }
    __syncthreads();

    // ---- per-anchor loss terms ----
    const float sumexp = ssum[tid];
    const long long g = blockAnchor0 + tid;
    const int b   = (int)(g / P_);            // blocks never straddle batch rows
    const int tgt = clsT[g];
    const float xt = scls[tid * C_ + tgt];
    const float ce = __logf(sumexp) + mx - xt;  // logsumexp - x[target]
    const bool pos = tgt > 0;

    const float4 lp = ((const float4*)locP)[g];
    const float4 lt = ((const float4*)locT)[g];
    float h = 0.0f;
    {
        const float dx[4] = {lp.x - lt.x, lp.y - lt.y, lp.z - lt.z, lp.w - lt.w};
        #pragma unroll
        for (int i = 0; i < 4; ++i) {
            const float ad = fabsf(dx[i]);
            h += (ad < 1.0f) ? 0.5f * dx[i] * dx[i] : (ad - 0.5f);
        }
    }

    float acc = pos ? (h + ce) : 0.0f;        // loc loss + positive CE
    (W + 128)[g] = pos ? 0.0f : fmaxf(ce, 0.0f);  // ce_neg row (>=0 keeps bit-order monotone)
    int np = pos ? 1 : 0;

    // wave32 reduction, then block combine + atomics
    #pragma unroll
    for (int off = 16; off > 0; off >>= 1) {
        acc += __shfl_xor(acc, off, 32);
        np  += __shfl_xor(np,  off, 32);
    }
    if (lane == 0) { sredf[tid >> 5] = acc; sredi[tid >> 5] = np; }
    __syncthreads();
    if (tid == 0) {
        const float a = sredf[0] + sredf[1] + sredf[2] + sredf[3];
        const int   n = sredi[0] + sredi[1] + sredi[2] + sredi[3];
        atomicAdd(W, a);
        atomicAdd(((int*)W) + 16 + b, n);
    }
}

// Exact per-row top-k sum via MSB-first radix select on f32 bit patterns (values >= 0).
__global__ __launch_bounds__(256)
void ssd_topk(float* __restrict__ W)
{
    __shared__ unsigned cnt[256];
    __shared__ float    fs[256];
    __shared__ unsigned sPrefix, sMask;

    const int b   = blockIdx.x;
    const int tid = threadIdx.x;
    const float* __restrict__ row = W + 128 + (size_t)b * P_;

    const int np   = ((const int*)W)[16 + b];
    const int negc = P_ - np;
    int k = 3 * np; if (k > negc) k = negc;
    if (k <= 0) return;                      // uniform per block

    if (tid == 0) { sPrefix = 0u; sMask = 0u; }
    __syncthreads();

    unsigned kRem = (unsigned)k;             // live on tid 0 only
    float    sumGt = 0.0f;

    for (int pass = 3; pass >= 0; --pass) {
        cnt[tid] = 0u; fs[tid] = 0.0f;
        __syncthreads();
        const unsigned pmask = sMask, pref = sPrefix;
        const int sh = pass * 8;
        for (int p = tid; p < P_; p += 256) {
            const float v = row[p];
            const unsigned u = __float_as_uint(v);
            if ((u & pmask) == pref) {
                const unsigned byte = (u >> sh) & 255u;
                atomicAdd(&cnt[byte], 1u);
                atomicAdd(&fs[byte], v);
            }
        }
        __syncthreads();
        if (tid == 0) {
            unsigned cum = 0; int sel = 0;
            for (int bin = 255; bin >= 0; --bin) {
                const unsigned c2 = cum + cnt[bin];
                if (c2 >= kRem) { sel = bin; break; }
                cum = c2; sumGt += fs[bin];
            }
            kRem -= cum;
            sPrefix = pref | ((unsigned)sel << sh);
            sMask   = pmask | (255u << sh);
        }
        __syncthreads();
    }
    if (tid == 0) {
        const float tv = __uint_as_float(sPrefix);   // exact k-th largest value
        atomicAdd(W, sumGt + (float)kRem * tv);      // ties contribute exactly
    }
}

__global__ void ssd_final(const float* __restrict__ W, float* __restrict__ out)
{
    __shared__ int s[64];
    const int t = threadIdx.x;
    s[t] = ((const int*)W)[16 + t];
    __syncthreads();
    if (t == 0) {
        int N = 0;
        for (int i = 0; i < 64; ++i) N += s[i];
        if (N < 1) N = 1;
        out[0] = W[0] / (float)N;
    }
}

extern "C" void kernel_launch(void* const* d_in, const int* in_sizes, int n_in,
                              void* d_out, int out_size, void* d_ws, size_t ws_size,
                              hipStream_t stream)
{
    const float* locP = (const float*)d_in[0];
    const float* clsP = (const float*)d_in[1];
    const float* locT = (const float*)d_in[2];
    const int*   clsT = (const int*)d_in[3];
    float* W   = (float*)d_ws;
    float* out = (float*)d_out;

    hipLaunchKernelGGL(ssd_zero,  dim3(1), dim3(128), 0, stream, W);
    hipLaunchKernelGGL(ssd_main,  dim3((B_ * P_) / BLK_ANCH), dim3(THREADS1), 0, stream,
                       locP, clsP, locT, clsT, W);
    hipLaunchKernelGGL(ssd_topk,  dim3(B_), dim3(256), 0, stream, W);
    hipLaunchKernelGGL(ssd_final, dim3(1), dim3(64), 0, stream, W, out);
}